// XDoGPass0Effect_56092272885899
// MI455X (gfx1250) — compile-verified
//
#include <hip/hip_runtime.h>
#include <stdint.h>

// XDoG pass-0: per-pixel directional double-Gaussian blur with bilinear taps.
// Strategy: stage a border-clamped 96x96 fp32 window per 32x32 output tile
// into LDS using the CDNA5 async global->LDS DMA path (ASYNCcnt): b128 DMA for
// interior tiles, clamped b32 DMA for border tiles. All 64 bilinear taps per
// pixel are then served from LDS with immediate-offset ds_loads (clamp is
// pre-baked into the staged window, so tap indexing is pure base+const).

namespace {
constexpr int   kW       = 1024;
constexpr int   kH       = 1024;
constexpr int   kTile    = 32;                  // output tile edge
constexpr int   kHalo    = 32;                  // rigorous: sqrt(2)*sqrt(15^2+16^2)+1 < 32.1
constexpr int   kReg     = kTile + 2 * kHalo;   // 96 staged window edge
constexpr int   kPitch   = 100;                 // row pitch (floats): 16B-aligned rows, conflict-safe
constexpr int   kThreads = 256;                 // 8 wave32
constexpr float kEps     = 1.1920928955078125e-07f; // jnp.finfo(f32).eps
constexpr float kRatio   = 1.6f;
constexpr float kLog2e   = 1.44269504088896340736f;
}

__device__ __forceinline__ float frcp(float x)  { return __builtin_amdgcn_rcpf(x); }
__device__ __forceinline__ float fexp2(float x) { return __builtin_amdgcn_exp2f(x); }

__global__ __launch_bounds__(kThreads)
void xdog_pass0_kernel(const float* __restrict__ xin,
                       const float* __restrict__ tangent,
                       const float* __restrict__ wkw,
                       const float* __restrict__ sigman,
                       float* __restrict__ out)
{
  __shared__ __align__(16) float tile[kReg * kPitch];

  const int hw = kH * kW;
  const int b  = blockIdx.z;
  const float* img = xin     + (size_t)b * hw;       // x[:,0]
  const float* tg0 = tangent + (size_t)b * 2 * hw;   // tangent[:,0]
  const float* tg1 = tg0 + hw;                       // tangent[:,1]
  const float* wk  = wkw    + (size_t)b * hw;
  const float* sg  = sigman + (size_t)b * hw;
  float*       op  = out    + (size_t)b * hw;

  const int tx0 = blockIdx.x * kTile;
  const int ty0 = blockIdx.y * kTile;
  const int gbx = tx0 - kHalo;
  const int gby = ty0 - kHalo;
  const int tid = threadIdx.y * 16 + threadIdx.x;

  // Warm the parameter lines while the DMA staging runs (global_prefetch_b8).
  __builtin_prefetch(&sg[(ty0 + (int)threadIdx.y) * kW + tx0 + (int)threadIdx.x], 0, 3);
  __builtin_prefetch(&tg0[(ty0 + (int)threadIdx.y) * kW + tx0 + (int)threadIdx.x], 0, 3);

  // ---- Stage the 96x96 window into LDS via async global->LDS DMA (ASYNCcnt).
  const bool interior = (gbx >= 0) & (gby >= 0) &
                        (gbx + kReg <= kW) & (gby + kReg <= kH);
  if (interior) {
    // Whole window in-bounds: 16B DMA. 96*24 = 2304 vec4 / 256 threads = 9 iters.
    for (int i = tid; i < kReg * (kReg / 4); i += kThreads) {
      const int ly = i / (kReg / 4);
      const int lx = (i - ly * (kReg / 4)) * 4;
      const uint64_t ga = (uint64_t)(uintptr_t)(img + ((gby + ly) * kW + gbx + lx));
      const uint32_t la = (uint32_t)(uintptr_t)(&tile[ly * kPitch + lx]);
      asm volatile("global_load_async_to_lds_b128 %0, %1, off"
                   :: "v"(la), "v"(ga) : "memory");
    }
  } else {
    // Border tile: element-wise DMA with border clamp baked into the window.
    for (int i = tid; i < kReg * kReg; i += kThreads) {
      const int ly = i / kReg;
      const int lx = i - ly * kReg;
      const int gx = min(max(gbx + lx, 0), kW - 1);
      const int gy = min(max(gby + ly, 0), kH - 1);
      const uint64_t ga = (uint64_t)(uintptr_t)(img + (gy * kW + gx));
      const uint32_t la = (uint32_t)(uintptr_t)(&tile[ly * kPitch + lx]);
      asm volatile("global_load_async_to_lds_b32 %0, %1, off"
                   :: "v"(la), "v"(ga) : "memory");
    }
  }
  asm volatile("s_wait_asynccnt 0x0" ::: "memory");
  __syncthreads();

  // Bilinear tap in tile coordinates. Border clamp is pre-baked into tile[],
  // and the halo proof guarantees ix,iy in [0, kReg-2], so no clamping here:
  // one base address + immediate ds offsets (+1, +pitch, +pitch+1).
  auto bilin = [&](float fx, float fy) -> float {
    const float flx = floorf(fx);
    const float fly = floorf(fy);
    const float wx  = fx - flx;
    const float wy  = fy - fly;
    const int base  = (int)fmaf(fly, (float)kPitch, flx);  // exact: < 2^24
    const float v00 = tile[base];
    const float v01 = tile[base + 1];
    const float v10 = tile[base + kPitch];
    const float v11 = tile[base + kPitch + 1];
    const float top = fmaf(wx, v01 - v00, v00);
    const float bot = fmaf(wx, v11 - v10, v10);
    return fmaf(wy, bot - top, top);
  };

  // Each thread computes a 2x2 scatter of output pixels within the 32x32 tile.
  for (int sub = 0; sub < 4; ++sub) {
    const int lxp = ((sub & 1) << 4) + (int)threadIdx.x;   // 0..31 in tile
    const int lyp = ((sub >> 1) << 4) + (int)threadIdx.y;
    const int px = tx0 + lxp;
    const int py = ty0 + lyp;
    const int pidx = py * kW + px;

    // grad_full = (tangent[1], -tangent[0])
    const float gfx = tg1[pidx];
    const float gfy = -tg0[pidx];
    const float me  = fmaxf(fabsf(gfx), fabsf(gfy));
    const float ds  = (me > 0.0f) ? (1.0f / me) : 1.0f;

    const float sn = sg[pidx];                 // * INPUT_SIZE_FACTOR (= 1.0)
    const float sw = sn * kRatio;
    const float half_step = 2.0f * sw;
    // exp(-q/(2s^2+eps)) == exp2(-q * log2e/(2s^2+eps)): fold log2e once.
    const float inv2sw = kLog2e * frcp(fmaf(2.0f * sw, sw, kEps));
    const float inv2sn = kLog2e * frcp(fmaf(2.0f * sn, sn, kEps));
    const float two_ds = 2.0f * ds;
    const float inv2ds = frcp(two_ds);
    const float reach  = half_step + two_ds;

    // Sample positions in tile coordinates (gbx/gby already subtracted).
    const float cx = (float)(kHalo + lxp);
    const float cy = (float)(kHalo + lyp);
    const float xc = img[pidx];

    float acc_w = 0.0f, acc_n = 0.0f, sum_w = 0.0f, sum_n = 0.0f;

    #pragma unroll
    for (int k = 0; k < 8; ++k) {
      const float n1 = (float)(2 * k + 1);
      const float n2 = (float)(2 * k + 2);
      const float i1 = ds * n1;
      const float i2 = ds * n2;
      const float q1 = i1 * i1;
      const float q2 = i2 * i2;
      // Gaussians are even in itr: shared across the +/- pair.
      const float gw1 = fexp2(-q1 * inv2sw);
      const float gn1 = fexp2(-q1 * inv2sn);
      const float gw2 = fexp2(-q2 * inv2sw);
      const float gn2 = fexp2(-q2 * inv2sn);
      const float gwide = gw1 + gw2;
      const float gnarr = gn1 + gn2;
      const float l1 = sqrtf(fmaf(gw1, gw1, fmaf(gn1, gn1, kEps)));
      const float l2 = sqrtf(fmaf(gw2, gw2, fmaf(gn2, gn2, kEps)));
      const float lk = sqrtf(fmaf(gwide, gwide, fmaf(gnarr, gnarr, kEps)));
      const float itr = (i1 * l1 + i2 * l2) * frcp(lk + kEps);

      const bool  in = (i1 <= reach);                       // |itr_d1| <= hs + 2ds
      const float dp = fminf(reach - i1, two_ds) * inv2ds;  // delta, + branch
      // - branch delta = 1 exactly (i1 >= 0). Mask/delta are shared by the
      // wide and narrow kernels, so fold them once:
      //   acc  += g_m * (tp*dp + tm),  norm += g_m * (dp + 1)
      const float gw_m = in ? gwide : 0.0f;
      const float gn_m = in ? gnarr : 0.0f;

      // fx = u*W - 0.5 = px + itr*grad_full_x (exact algebra of the reference)
      const float ox = itr * gfx;
      const float oy = itr * gfy;
      const float tp = bilin(cx + ox, cy + oy);
      const float tm = bilin(cx - ox, cy - oy);

      const float c = fmaf(tp, dp, tm);
      const float s = dp + 1.0f;
      acc_w = fmaf(gw_m, c, acc_w);
      acc_n = fmaf(gn_m, c, acc_n);
      sum_w = fmaf(gw_m, s, sum_w);
      sum_n = fmaf(gn_m, s, sum_n);
    }

    const float rw = (acc_w + xc) * frcp(sum_w + 1.0f);
    const float rn = (acc_n + xc) * frcp(sum_n + 1.0f);
    const float p  = 100.0f * wk[pidx];
    op[pidx] = (1.0f + p) * rn - p * rw;
  }
}

extern "C" void kernel_launch(void* const* d_in, const int* in_sizes, int n_in,
                              void* d_out, int out_size, void* d_ws, size_t ws_size,
                              hipStream_t stream) {
  const float* x   = (const float*)d_in[0];
  const float* tg  = (const float*)d_in[1];
  const float* wkw = (const float*)d_in[2];
  const float* sn  = (const float*)d_in[3];
  float* out = (float*)d_out;

  int B = in_sizes[0] / (kW * kH);
  if (B < 1) B = 1;
  dim3 grid(kW / kTile, kH / kTile, B);
  dim3 block(16, 16, 1);
  hipLaunchKernelGGL(xdog_pass0_kernel, grid, block, 0, stream,
                     x, tg, wkw, sn, out);
}